// GraphCon_74990128988566
// MI455X (gfx1250) — compile-verified
//
#include <hip/hip_runtime.h>
#include <hip/hip_bf16.h>

typedef __attribute__((ext_vector_type(2))) float v2f;
typedef __attribute__((ext_vector_type(8))) float v8f;

#define F_IN   128
#define UNITS  128

// ---------------------------------------------------------------------------
// Kernel 1: zero the aggregation scratch (float4 stores).
// ---------------------------------------------------------------------------
__global__ __launch_bounds__(256) void gc_zero_kernel(float4* __restrict__ p, int n4) {
    int i = blockIdx.x * blockDim.x + threadIdx.x;
    if (i < n4) p[i] = make_float4(0.f, 0.f, 0.f, 0.f);
}

// ---------------------------------------------------------------------------
// Kernel 2: SpMM scatter.  One wave32 per edge; lane l handles features
// [4l, 4l+4): coalesced float4 gather of features[src], scaled by edge val,
// scatter-added into agg[dst] with global f32 atomics (L2-resident, 51.2MB).
// ---------------------------------------------------------------------------
__global__ __launch_bounds__(256) void gc_scatter_kernel(
    const float*  __restrict__ features,
    const int*    __restrict__ edge_src,
    const int*    __restrict__ edge_dst,
    const float*  __restrict__ edge_vals,
    float*        __restrict__ agg,
    int n_edges)
{
    int gid  = blockIdx.x * blockDim.x + threadIdx.x;
    int e    = gid >> 5;
    int lane = gid & 31;
    if (e >= n_edges) return;

    int   s = edge_src[e];
    int   d = edge_dst[e];
    float v = edge_vals[e];

    const float4 f = ((const float4*)(features + (size_t)s * F_IN))[lane];
    float* ap = agg + (size_t)d * F_IN + lane * 4;
    atomicAdd(ap + 0, f.x * v);
    atomicAdd(ap + 1, f.y * v);
    atomicAdd(ap + 2, f.z * v);
    atomicAdd(ap + 3, f.w * v);
}

// ---------------------------------------------------------------------------
// Kernel 3: out = relu(agg @ W + bias), fp32 WMMA (V_WMMA_F32_16X16X4_F32).
//
// Block = 256 threads (8 waves).  Block owns a 16-row strip of agg; wave w
// owns output column tile [16w, 16w+16).  W (128x128, 64KB) and the A strip
// (16x128, 8KB) are staged in LDS.  K-loop: 32 WMMA steps of K=4.
//
// Fragment layouts (CDNA5 ISA 7.12.2, wave32):
//   A (16x4 f32, 2 VGPR):  M = lane%16, K = 2*(lane/16) + {vgpr0, vgpr1}
//   B (4x16 f32, 2 VGPR):  N = lane%16, K = 2*(lane/16) + {vgpr0, vgpr1}
//   C/D (16x16 f32, 8 VGPR): VGPR r -> M = r + 8*(lane/16), N = lane%16
// ---------------------------------------------------------------------------
__global__ __launch_bounds__(256) void gc_gemm_bias_relu_kernel(
    const float* __restrict__ agg,
    const float* __restrict__ W,
    const float* __restrict__ bias,
    float*       __restrict__ out,
    int n_rows)
{
    __shared__ float sW[F_IN * UNITS];   // 64 KB
    __shared__ float sA[16 * F_IN];      //  8 KB

    const int tid  = threadIdx.x;
    const int row0 = blockIdx.x * 16;

    // Stage W: 16384 floats = 4096 float4, 16 per thread.
    {
        const float4* W4  = (const float4*)W;
        float4*       sW4 = (float4*)sW;
#pragma unroll
        for (int i = 0; i < 16; ++i)
            sW4[tid + i * 256] = W4[tid + i * 256];
    }
    // Stage A strip: 2048 floats = 512 float4, 2 per thread (row-clamped).
    {
        float4* sA4 = (float4*)sA;
#pragma unroll
        for (int i = 0; i < 2; ++i) {
            int ei  = tid + i * 256;          // float4 index within strip
            int r   = ei >> 5;                // 32 float4 per row
            int c4  = ei & 31;
            int gr  = row0 + r;
            if (gr >= n_rows) gr = n_rows - 1;
            sA4[ei] = ((const float4*)(agg + (size_t)gr * F_IN))[c4];
        }
    }
    __syncthreads();

    const int wave = tid >> 5;
    const int lane = tid & 31;
    const int n0   = wave * 16;            // this wave's output column tile
    const int mn   = lane & 15;            // M for A, N for B/C/D
    const int kgrp = (lane >> 4) << 1;     // 0 or 2

    v8f c = {};
#pragma unroll
    for (int k0 = 0; k0 < F_IN; k0 += 4) {
        const int kb = k0 + kgrp;
        v2f a, b;
        a.x = sA[mn * F_IN + kb];
        a.y = sA[mn * F_IN + kb + 1];
        b.x = sW[kb * UNITS + n0 + mn];
        b.y = sW[(kb + 1) * UNITS + n0 + mn];
        c = __builtin_amdgcn_wmma_f32_16x16x4_f32(
                /*neg_a=*/false, a, /*neg_b=*/false, b,
                /*c_mod=*/(short)0, c, /*reuse_a=*/false, /*reuse_b=*/false);
    }

    // Bias + ReLU + store (divergence here is fine; WMMA loop is done).
    const int mbase = row0 + ((lane >> 4) << 3);
    const int ncol  = n0 + mn;
    const float bv  = bias[ncol];
#pragma unroll
    for (int r = 0; r < 8; ++r) {
        int grow = mbase + r;
        if (grow < n_rows) {
            float v = c[r] + bv;
            out[(size_t)grow * UNITS + ncol] = v > 0.f ? v : 0.f;
        }
    }
}

// ---------------------------------------------------------------------------
// Host-side launcher.
// Inputs (setup_inputs order): features [N,128] f32, edge_src [E] i32,
// edge_dst [E] i32, edge_vals [E] f32, kernel [128,128] f32, bias [128] f32.
// Output: [N,128] f32.  Workspace: agg [N,128] f32 (51.2 MB).
// ---------------------------------------------------------------------------
extern "C" void kernel_launch(void* const* d_in, const int* in_sizes, int n_in,
                              void* d_out, int out_size, void* d_ws, size_t ws_size,
                              hipStream_t stream) {
    const float* features  = (const float*)d_in[0];
    const int*   edge_src  = (const int*)  d_in[1];
    const int*   edge_dst  = (const int*)  d_in[2];
    const float* edge_vals = (const float*)d_in[3];
    const float* Wk        = (const float*)d_in[4];
    const float* bias      = (const float*)d_in[5];
    float*       out       = (float*)d_out;
    float*       agg       = (float*)d_ws;

    const int N = in_sizes[0] / F_IN;
    const int E = in_sizes[1];

    // 1. Zero agg (must happen every call; ws is not re-poisoned).
    {
        int n4   = (N * F_IN) / 4;
        int grid = (n4 + 255) / 256;
        gc_zero_kernel<<<grid, 256, 0, stream>>>((float4*)agg, n4);
    }
    // 2. Scatter-add messages.
    {
        long long threads = (long long)E * 32;
        int grid = (int)((threads + 255) / 256);
        gc_scatter_kernel<<<grid, 256, 0, stream>>>(
            features, edge_src, edge_dst, edge_vals, agg, E);
    }
    // 3. WMMA GEMM + bias + relu.
    {
        int grid = (N + 15) / 16;   // N=100000 -> exactly 6250 full tiles
        gc_gemm_bias_relu_kernel<<<grid, 256, 0, stream>>>(
            agg, Wk, bias, out, N);
    }
}